// MoNet_85856396247842
// MI455X (gfx1250) — compile-verified
//
#include <hip/hip_runtime.h>
#include <math.h>

// Problem constants (match reference)
#define NNODES 50000
#define NEDGES 800000
#define INFEAT 128
#define HID    64
#define OUTF   40
#define KKER   3
#define KH0    (KKER * HID)   // 192
#define KH1    (KKER * OUTF)  // 120

typedef __attribute__((ext_vector_type(2))) float v2f;
typedef __attribute__((ext_vector_type(8))) float v8f;

// ---------------------------------------------------------------------------
// Degree / pseudo-coordinate prep
// ---------------------------------------------------------------------------
__global__ void zero_f32(float* __restrict__ p, int n) {
  int i = blockIdx.x * blockDim.x + threadIdx.x;
  if (i < n) p[i] = 0.0f;
}

__global__ void count_degree(const int* __restrict__ dst, float* __restrict__ deg, int e) {
  int i = blockIdx.x * blockDim.x + threadIdx.x;
  if (i < e) atomicAdd(&deg[dst[i]], 1.0f);
}

__global__ void inv_sqrt_deg(const float* __restrict__ deg, float* __restrict__ isd, int n) {
  int i = blockIdx.x * blockDim.x + threadIdx.x;
  if (i < n) isd[i] = rsqrtf(deg[i]);
}

__global__ void init_bias(float* __restrict__ out, const float* __restrict__ b, int n, int ch) {
  int i = blockIdx.x * blockDim.x + threadIdx.x;
  if (i < n * ch) out[i] = b[i % ch];
}

// ---------------------------------------------------------------------------
// fp32 GEMM via V_WMMA_F32_16X16X4_F32.
// C[M x Ncols] = A[M x Kdim] @ B[Kdim x Ncols], all row-major fp32.
// Block = 128 threads = 4 waves; wave w owns a PAIR of adjacent 16x16 output
// tiles (columns [col0, col0+32)) sharing one A fragment per K-step:
//   per K-step: 1x global_load_b64 (A) + 4x b32 (B) + 2x v_wmma.
// Requires M % 16 == 0 (true: 50000 = 3125*16) and Kdim % 4 == 0.
// Ragged Ncols: B column indices are CLAMPED in-bounds (no predication);
// garbage contributions land only in output columns the guarded store skips
// (D column n depends only on B column n).
//
// ISA lane layouts (CDNA5 ISA):
//  A 16x4:  lanes 0-15 -> M=lane, VGPR0=K0, VGPR1=K1 ; lanes 16-31 -> K2,K3
//  B 4x16:  row striped across lanes within a VGPR (K halves mirror A)
//  C/D:     lanes 0-15 -> N=lane, rows 0..7 in v[0..7]; lanes 16-31 -> rows 8..15
// ---------------------------------------------------------------------------
__global__ __launch_bounds__(128) void gemm_wmma_f32(
    const float* __restrict__ A, const float* __restrict__ B, float* __restrict__ C,
    int M, int Kdim, int Ncols) {
  const int wave = threadIdx.x >> 5;
  const int lane = threadIdx.x & 31;
  const int half = lane >> 4;   // 0: lanes 0-15, 1: lanes 16-31
  const int sub  = lane & 15;

  const int row0 = blockIdx.y * 16;
  const int col0 = (blockIdx.x * 4 + wave) * 32;   // two tiles: [col0, col0+32)
  if (col0 >= Ncols) return;

  const int koff = half * 2;  // this lane covers K = kk+koff, kk+koff+1
  // Clamp B columns in-bounds; out-of-range columns are never stored.
  const int bcol0 = min(col0 + sub,      Ncols - 1);
  const int bcol1 = min(col0 + 16 + sub, Ncols - 1);
  const float* arow  = A + (size_t)(row0 + sub) * Kdim + koff;
  const float* bbase = B + (size_t)koff * Ncols;

  v8f acc0 = {};
  v8f acc1 = {};
  for (int kk = 0; kk < Kdim; kk += 4) {
    v2f a, b0, b1;
    a.x = arow[kk];
    a.y = arow[kk + 1];
    const float* brow = bbase + (size_t)kk * Ncols;
    b0.x = brow[bcol0];
    b0.y = brow[(size_t)Ncols + bcol0];
    b1.x = brow[bcol1];
    b1.y = brow[(size_t)Ncols + bcol1];
    // 8 args: (neg_a, A, neg_b, B, c_mod, C, reuse_a, reuse_b)
    acc0 = __builtin_amdgcn_wmma_f32_16x16x4_f32(
        false, a, false, b0, (short)0, acc0, false, false);
    acc1 = __builtin_amdgcn_wmma_f32_16x16x4_f32(
        false, a, false, b1, (short)0, acc1, false, false);
  }

  // Store D tiles: lane half selects row group 0..7 / 8..15.
  const int crow0 = row0 + half * 8;
  const int c0 = col0 + sub;
  if (c0 < Ncols) {
#pragma unroll
    for (int r = 0; r < 8; ++r) {
      C[(size_t)(crow0 + r) * Ncols + c0] = acc0[r];
    }
  }
  const int c1 = col0 + 16 + sub;
  if (c1 < Ncols) {
#pragma unroll
    for (int r = 0; r < 8; ++r) {
      C[(size_t)(crow0 + r) * Ncols + c1] = acc1[r];
    }
  }
}

// ---------------------------------------------------------------------------
// Edge aggregation: one wave per edge.
// pd   = tanh([isd[src], isd[dst]] @ Wp + bp)                (DIM = 2)
// g[k] = exp(-0.5 * sum_d ((pd[d]-mu[k,d]) * isig[k,d])^2)   (K = 3)
// out[dst, c] += sum_k g[k] * hp[src, k, c]    for c in [0, CH)
// hp row stride = 3*CH. Scatter via global fp32 atomics (resolved in L2;
// hp0/hp1/h/out all fit in the 192MB L2, so gathers/atomics stay on-chip).
// ---------------------------------------------------------------------------
__global__ __launch_bounds__(256) void edge_aggregate(
    const int* __restrict__ src, const int* __restrict__ dst,
    const float* __restrict__ isd,
    const float* __restrict__ Wp, const float* __restrict__ bp,
    const float* __restrict__ mu, const float* __restrict__ isig,
    const float* __restrict__ hp, float* __restrict__ out,
    int e_total, int CH) {
  const int wavesPerBlock = blockDim.x >> 5;
  const int e = blockIdx.x * wavesPerBlock + (threadIdx.x >> 5);
  if (e >= e_total) return;
  const int lane = threadIdx.x & 31;

  const int s = src[e];
  const int d = dst[e];
  const float p0 = isd[s];
  const float p1 = isd[d];

  // Linear(2, 2) + tanh; Wp is row-major (2 x DIM)
  const float pd0 = tanhf(p0 * Wp[0] + p1 * Wp[2] + bp[0]);
  const float pd1 = tanhf(p0 * Wp[1] + p1 * Wp[3] + bp[1]);

  float g[KKER];
#pragma unroll
  for (int k = 0; k < KKER; ++k) {
    const float dx = (pd0 - mu[k * 2 + 0]) * isig[k * 2 + 0];
    const float dy = (pd1 - mu[k * 2 + 1]) * isig[k * 2 + 1];
    g[k] = __expf(-0.5f * (dx * dx + dy * dy));
  }

  const float* hs = hp + (size_t)s * (3 * CH);
  float* od = out + (size_t)d * CH;
  for (int c = lane; c < CH; c += 32) {
    const float acc = g[0] * hs[c] + g[1] * hs[CH + c] + g[2] * hs[2 * CH + c];
    atomicAdd(&od[c], acc);
  }
}

// ---------------------------------------------------------------------------
// Launcher
// ---------------------------------------------------------------------------
extern "C" void kernel_launch(void* const* d_in, const int* in_sizes, int n_in,
                              void* d_out, int out_size, void* d_ws, size_t ws_size,
                              hipStream_t stream) {
  (void)in_sizes; (void)n_in; (void)out_size; (void)ws_size;

  const float* feat  = (const float*)d_in[0];
  const int*   src   = (const int*)  d_in[1];
  const int*   dst   = (const int*)  d_in[2];
  const float* Wp0   = (const float*)d_in[3];
  const float* bp0   = (const float*)d_in[4];
  const float* mu0   = (const float*)d_in[5];
  const float* isig0 = (const float*)d_in[6];
  const float* Wfc0  = (const float*)d_in[7];
  const float* b0    = (const float*)d_in[8];
  const float* Wp1   = (const float*)d_in[9];
  const float* bp1   = (const float*)d_in[10];
  const float* mu1   = (const float*)d_in[11];
  const float* isig1 = (const float*)d_in[12];
  const float* Wfc1  = (const float*)d_in[13];
  const float* b1    = (const float*)d_in[14];

  // Workspace layout (fp32): deg | isd | hp0[N,192] | h[N,64] | hp1[N,120]
  float* deg = (float*)d_ws;
  float* isd = deg + NNODES;
  float* hp0 = isd + NNODES;
  float* h   = hp0 + (size_t)NNODES * KH0;
  float* hp1 = h   + (size_t)NNODES * HID;
  float* out = (float*)d_out;

  const int T = 256;

  // 1) in-degree -> 1/sqrt(deg)
  zero_f32<<<(NNODES + T - 1) / T, T, 0, stream>>>(deg, NNODES);
  count_degree<<<(NEDGES + T - 1) / T, T, 0, stream>>>(dst, deg, NEDGES);
  inv_sqrt_deg<<<(NNODES + T - 1) / T, T, 0, stream>>>(deg, isd, NNODES);

  // 2) hp0 = feat @ Wfc0  (50000x128 @ 128x192); 6 col-pairs -> 2 blocks x 4 waves
  {
    dim3 grid(2, NNODES / 16);
    gemm_wmma_f32<<<grid, 128, 0, stream>>>(feat, Wfc0, hp0, NNODES, INFEAT, KH0);
  }

  // 3) h = b0; scatter layer-0 messages
  init_bias<<<(NNODES * HID + T - 1) / T, T, 0, stream>>>(h, b0, NNODES, HID);
  {
    const int wavesPerBlock = T / 32;  // 8 edges per block
    const int blocks = (NEDGES + wavesPerBlock - 1) / wavesPerBlock;
    edge_aggregate<<<blocks, T, 0, stream>>>(src, dst, isd, Wp0, bp0, mu0, isig0,
                                             hp0, h, NEDGES, HID);
  }

  // 4) hp1 = h @ Wfc1  (50000x64 @ 64x120); 4 col-pairs -> 1 block x 4 waves
  {
    dim3 grid(1, NNODES / 16);
    gemm_wmma_f32<<<grid, 128, 0, stream>>>(h, Wfc1, hp1, NNODES, HID, KH1);
  }

  // 5) out = b1; scatter layer-1 messages
  init_bias<<<(NNODES * OUTF + T - 1) / T, T, 0, stream>>>(out, b1, NNODES, OUTF);
  {
    const int wavesPerBlock = T / 32;
    const int blocks = (NEDGES + wavesPerBlock - 1) / wavesPerBlock;
    edge_aggregate<<<blocks, T, 0, stream>>>(src, dst, isd, Wp1, bp1, mu1, isig1,
                                             hp1, out, NEDGES, OUTF);
  }
}